// TMPIRenderer_62345745269074
// MI455X (gfx1250) — compile-verified
//
#include <hip/hip_runtime.h>

// Problem constants (from the reference)
#define S_PLANES 512
#define HP_DIM   144
#define P_ELEM   20736      // 144*144
#define RGB_STRIDE 62208    // 3*P_ELEM
#define OUT_W    512
#define OUT_H    512
#define OUT_PIX  262144     // 512*512
#define CROP_PAD 8
#define BLK_W    128        // block pixel footprint: 128 x 8
#define BLK_H    8
#define PF_LA    16         // prefetch lookahead (list entries)

static __device__ __forceinline__ unsigned ballot32(bool p) {
#if __has_builtin(__builtin_amdgcn_ballot_w32)
  return __builtin_amdgcn_ballot_w32(p);
#else
  return (unsigned)__ballot((int)p);
#endif
}

// Align-4 16-byte payload: tile rows are arbitrarily offset, so only 4B
// alignment is guaranteed. Probe-confirmed (round 4): this lowers to a single
// global_load_b128 on gfx1250 (unaligned-access mode), no splitting.
struct __attribute__((packed, aligned(4))) F4 { float x, y, z, w; };
// Output stores are genuinely 16B-aligned.
struct __attribute__((aligned(16))) F4A { float x, y, z, w; };

// ---------------------------------------------------------------------------
// Kernel: front-to-back compositing, 4 pixels per thread (b128 streams).
// Block covers a 128x8 pixel footprint; wave 0 ballot-compacts (order
// preserving) the planes whose tile intersects the footprint into LDS,
// then all 8 waves scan only that list. global_prefetch_b8 lookahead keeps
// ~18KB/SIMD of lines in flight despite the serial transmittance chain.
// ---------------------------------------------------------------------------
__global__ __launch_bounds__(256) void tmpi_composite(
    const float* __restrict__ rgb,     // (S,3,P)
    const float* __restrict__ sigma,   // (S,P)
    const unsigned* __restrict__ meta, // (S) depth-sorted packed
    float* __restrict__ out) {         // (3,512,512)
  __shared__ unsigned s_list[S_PLANES];
  __shared__ int s_n;

  const int tid = threadIdx.x;
  const int cx0 = blockIdx.x * BLK_W + CROP_PAD;  // block canvas col base
  const int cy0 = blockIdx.y * BLK_H + CROP_PAD;  // block canvas row base

  // ---- per-block plane culling (wave 0 only), depth order preserved ----
  if (tid < 32) {
    const int lane = tid;
    int cnt = 0;
    const unsigned lt_mask = (1u << lane) - 1u;
    for (int c = 0; c < S_PLANES / 32; ++c) {
      const unsigned m = meta[c * 32 + lane];
      const int psx = (int)(m & 1023u);
      const int psy = (int)((m >> 10) & 1023u);
      const int dx = psx - cx0 + (HP_DIM - 1);
      const int dy = psy - cy0 + (HP_DIM - 1);
      const bool hit = ((unsigned)dx < (unsigned)(HP_DIM + BLK_W - 1)) &
                       ((unsigned)dy < (unsigned)(HP_DIM + BLK_H - 1));
      const unsigned msk = ballot32(hit);
      if (hit) s_list[cnt + __popc(msk & lt_mask)] = m;
      cnt += __popc(msk);
    }
    if (lane == 0) s_n = cnt;
  }
  __syncthreads();

  // ---- per-thread: 4 consecutive x-pixels, one y-row ----
  const int gx   = tid & 31;                 // 4-pixel group within row
  const int row  = tid >> 5;                 // 0..7
  const int cx0g = cx0 + 4 * gx;             // canvas x of pixel 0 of group
  const int cy   = cy0 + row;

  const int n  = s_n;
  const int nf = n - PF_LA;                  // prefetch-eligible bound
  float T[4]  = {1.0f, 1.0f, 1.0f, 1.0f};
  float aR[4] = {0.0f, 0.0f, 0.0f, 0.0f};
  float aG[4] = {0.0f, 0.0f, 0.0f, 0.0f};
  float aB[4] = {0.0f, 0.0f, 0.0f, 0.0f};

#define STEP(j, sv, rv, gv, bv)                       \
  {                                                   \
    const float w_ = (sv) * T[j];                     \
    aR[j] = fmaf((rv), w_, aR[j]);                    \
    aG[j] = fmaf((gv), w_, aG[j]);                    \
    aB[j] = fmaf((bv), w_, aB[j]);                    \
    T[j]  = fmaf(-(sv), T[j], T[j]);                  \
  }

  for (int e = 0; e < n; ++e) {
    const unsigned m =
        (unsigned)__builtin_amdgcn_readfirstlane((int)s_list[e]);

    // ---- prefetch lookahead: warm GL2/WGP$ ahead of the serial T chain ----
    if (e < nf) {
      const unsigned mf =
          (unsigned)__builtin_amdgcn_readfirstlane((int)s_list[e + PF_LA]);
      const int fy = cy - (int)((mf >> 10) & 1023u);
      int fx = cx0g - (int)(mf & 1023u);
      if (((unsigned)fy < (unsigned)HP_DIM) & ((unsigned)(fx + 3) < 147u)) {
        if (fx < 0) fx = 0;
        const int fid = (int)(mf >> 20);
        const int flp = fy * HP_DIM + fx;
        const float* fsb = sigma + (size_t)fid * P_ELEM + flp;
        const float* frb = rgb + (size_t)fid * RGB_STRIDE + flp;
        __builtin_prefetch(fsb, 0, 1);
        __builtin_prefetch(frb, 0, 1);
        __builtin_prefetch(frb + P_ELEM, 0, 1);
        __builtin_prefetch(frb + 2 * P_ELEM, 0, 1);
      }
    }

    // ---- composite current plane ----
    const int psx = (int)(m & 1023u);
    const int psy = (int)((m >> 10) & 1023u);
    const int id  = (int)(m >> 20);
    const int ly  = cy - psy;
    const int lx0 = cx0g - psx;
    if ((unsigned)ly < (unsigned)HP_DIM) {
      const float* sb = sigma + (size_t)id * P_ELEM;
      const float* rb = rgb   + (size_t)id * RGB_STRIDE;
      if ((unsigned)lx0 < 141u) {
        // group fully inside the tile: single-b128 streams (probe-confirmed)
        const int lp = ly * HP_DIM + lx0;
        const F4 s4 = *(const F4*)(sb + lp);
        const F4 r4 = *(const F4*)(rb + lp);
        const F4 g4 = *(const F4*)(rb + lp + P_ELEM);
        const F4 b4 = *(const F4*)(rb + lp + 2 * P_ELEM);
        STEP(0, s4.x, r4.x, g4.x, b4.x);
        STEP(1, s4.y, r4.y, g4.y, b4.y);
        STEP(2, s4.z, r4.z, g4.z, b4.z);
        STEP(3, s4.w, r4.w, g4.w, b4.w);
      } else if ((unsigned)(lx0 + 3) < 147u) {
        // group straddles a tile edge: per-pixel fallback (no OOB reads)
#pragma unroll
        for (int j = 0; j < 4; ++j) {
          const int lx = lx0 + j;
          if ((unsigned)lx < (unsigned)HP_DIM) {
            const int lp = ly * HP_DIM + lx;
            const float s  = sb[lp];
            const float r0 = rb[lp];
            const float g0 = rb[lp + P_ELEM];
            const float b0 = rb[lp + 2 * P_ELEM];
            STEP(j, s, r0, g0, b0);
          }
        }
      }
    }
  }
#undef STEP

  // ---- store 4 consecutive pixels per channel as b128 (16B-aligned) ----
  const int ox = blockIdx.x * BLK_W + 4 * gx;
  const int oy = blockIdx.y * BLK_H + row;
  const int o  = oy * OUT_W + ox;           // multiple of 4 -> 16B aligned
  *(F4A*)(out + o)               = F4A{aR[0], aR[1], aR[2], aR[3]};
  *(F4A*)(out + OUT_PIX + o)     = F4A{aG[0], aG[1], aG[2], aG[3]};
  *(F4A*)(out + 2 * OUT_PIX + o) = F4A{aB[0], aB[1], aB[2], aB[3]};
}

// ---------------------------------------------------------------------------
// Kernel: single-block bitonic sort of 512 (depth, index) pairs, ascending.
// Emits packed per-plane metadata in depth order: sx | sy<<10 | planeIdx<<20.
// ---------------------------------------------------------------------------
__global__ __launch_bounds__(256) void tmpi_sort_planes(
    const float* __restrict__ depth,
    const int*   __restrict__ sx,
    const int*   __restrict__ sy,
    unsigned*    __restrict__ meta) {
  __shared__ float skey[S_PLANES];
  __shared__ int   sval[S_PLANES];
  const int t = threadIdx.x;

  for (int i = t; i < S_PLANES; i += 256) { skey[i] = depth[i]; sval[i] = i; }
  __syncthreads();

  for (int k = 2; k <= S_PLANES; k <<= 1) {
    for (int j = k >> 1; j > 0; j >>= 1) {
      for (int i = t; i < S_PLANES; i += 256) {
        const int ixj = i ^ j;
        if (ixj > i) {
          const float a = skey[i];
          const float b = skey[ixj];
          const bool asc = (i & k) == 0;
          if (asc ? (a > b) : (a < b)) {
            skey[i] = b; skey[ixj] = a;
            const int tv = sval[i]; sval[i] = sval[ixj]; sval[ixj] = tv;
          }
        }
      }
      __syncthreads();
    }
  }

  for (int i = t; i < S_PLANES; i += 256) {
    const int id = sval[i];
    meta[i] = (unsigned)sx[id] | ((unsigned)sy[id] << 10) | ((unsigned)id << 20);
  }
}

// ---------------------------------------------------------------------------
// Host-side launcher
// ---------------------------------------------------------------------------
extern "C" void kernel_launch(void* const* d_in, const int* in_sizes, int n_in,
                              void* d_out, int out_size, void* d_ws, size_t ws_size,
                              hipStream_t stream) {
  const float* rgb   = (const float*)d_in[0];  // (1,512,3,20736)
  const float* sigma = (const float*)d_in[1];  // (1,512,20736)
  const float* depth = (const float*)d_in[2];  // (1,512)
  const int*   sx    = (const int*)d_in[3];    // (1,512)
  const int*   sy    = (const int*)d_in[4];    // (1,512)
  float* out = (float*)d_out;                  // (1,3,512,512)
  unsigned* meta = (unsigned*)d_ws;            // 512 * 4 bytes of scratch

  tmpi_sort_planes<<<dim3(1), dim3(256), 0, stream>>>(depth, sx, sy, meta);
  tmpi_composite<<<dim3(OUT_W / BLK_W, OUT_H / BLK_H), dim3(256), 0, stream>>>(
      rgb, sigma, meta, out);
}